// Capsule_82033875353735
// MI455X (gfx1250) — compile-verified
//
#include <hip/hip_runtime.h>
#include <hip/hip_bf16.h>
#include <math.h>

typedef __attribute__((ext_vector_type(2))) float v2f;
typedef __attribute__((ext_vector_type(8))) float v8f;

#define B_TOT  256
#define N_CAPS 1152
#define DIN    8
#define C_CAPS 10
#define U_DIM  16
#define BT     16               // batch tile per workgroup
#define WAVES  8
#define NPW    (N_CAPS / WAVES) // 144 n-values per wave

// Transposed u_hat tile for fixed (c, n):
//   D'[u][m] += sum_i W[n,i,u] * X[b0+m][n][i]   (= u_hat[m,n,u])
// A = Wn^T (M=u, K=i), B = Xn^T (K=i, N=m), two K=4 WMMAs for i=0..7.
// xrow: lane-specific &X[b0 + (lane&15)][n][kb]   (kb = (lane>>4)*2)
// wrow: lane-specific &W[c][n][kb][lane&15]
__device__ __forceinline__ v8f uhatT_tile(const float* __restrict__ xrow,
                                          const float* __restrict__ wrow,
                                          v8f acc) {
  v2f wa1 = { wrow[0],  wrow[16] };   // A: M=u=lane&15, K=i=kb,kb+1
  v2f wa2 = { wrow[64], wrow[80] };   // A: K=i=kb+4,kb+5
  v2f xb1 = { xrow[0],  xrow[1]  };   // B: N=m=lane&15, K=i=kb,kb+1
  v2f xb2 = { xrow[4],  xrow[5]  };   // B: K=i=kb+4,kb+5
  acc = __builtin_amdgcn_wmma_f32_16x16x4_f32(false, wa1, false, xb1,
                                              (short)0, acc, false, false);
  acc = __builtin_amdgcn_wmma_f32_16x16x4_f32(false, wa2, false, xb2,
                                              (short)0, acc, false, false);
  return acc;
}

__global__ __launch_bounds__(256)
void caps_route_kernel(const float* __restrict__ X,    // (256,1152,8)
                       const float* __restrict__ W,    // (10,1152,8,16)
                       float* __restrict__ out) {      // (256,10,16)
  __shared__ float sred[WAVES * BT * U_DIM];  // cross-wave partials (8 KB)
  __shared__ float svec[BT * U_DIM];          // s then v (m-major)
  __shared__ float vcum[BT * U_DIM];          // cumulative sum of v's (m-major)
  __shared__ float rowsuminv[BT];
  __shared__ float rowscale[BT];

  const int tid  = threadIdx.x;
  const int wv   = tid >> 5;
  const int lane = tid & 31;
  const int c    = blockIdx.x >> 4;            // 0..9
  const int bt0  = (blockIdx.x & 15) * BT;     // batch tile base

  const int mcol = lane & 15;                  // m column (and u index base)
  const int kb   = (lane >> 4) * 2;            // K base within half-wave
  const int hi8  = (lane >> 4) * 8;            // D' u-row offset for this half

  const float* Xb = X + (size_t)bt0 * N_CAPS * DIN;
  const float* Wc = W + (size_t)c * N_CAPS * DIN * U_DIM;

  // lane-fixed base pointers; advance with n
  const float* xbase = Xb + (size_t)mcol * N_CAPS * DIN + kb;   // += 8 per n
  const float* wbase = Wc + (size_t)kb * U_DIM + mcol;          // += 128 per n

  if (tid < BT * U_DIM) vcum[tid] = 0.0f;
  __syncthreads();

  const int n0 = wv * NPW;

  for (int iter = 0; iter < 3; ++iter) {
    float vrow[8];          // Vcum[m = mcol][u = hi8 + j]
    if (iter > 0) {
      #pragma unroll
      for (int j = 0; j < 8; ++j) vrow[j] = vcum[mcol * U_DIM + hi8 + j];
    }

    // ---- stats pass: rowsum[m] = sum_n exp(1 + uhat[m,n,:].Vcum[m,:]) ----
    if (iter > 0) {
      float psum = 0.0f;
      const float* xp = xbase + (size_t)n0 * DIN;
      const float* wp = wbase + (size_t)n0 * DIN * U_DIM;
      for (int nn = 0; nn < NPW; ++nn) {
        __builtin_prefetch(wp + 4 * DIN * U_DIM, 0, 1);
        v8f d = {0.f,0.f,0.f,0.f,0.f,0.f,0.f,0.f};
        d = uhatT_tile(xp, wp, d);
        float q = 0.0f;
        #pragma unroll
        for (int j = 0; j < 8; ++j) q = fmaf(d[j], vrow[j], q);
        q += __shfl_xor(q, 16);               // combine u=0..7 and u=8..15 halves
        psum += __expf(1.0f + q);             // b = 1 + uhat.Vcum
        xp += DIN; wp += DIN * U_DIM;
      }
      if (lane < 16) sred[wv * 16 + lane] = psum;   // lane owns m = lane
      __syncthreads();
      if (tid < BT) {
        float s = 0.f;
        for (int w = 0; w < WAVES; ++w) s += sred[w * 16 + tid];
        rowsuminv[tid] = 1.0f / s;
      }
      __syncthreads();
    }

    // ---- s pass: sT[u,m] = sum_n coeff[m,n] * uhatT[u,m] ----
    float sacc[8] = {0.f,0.f,0.f,0.f,0.f,0.f,0.f,0.f};
    {
      const float* xp = xbase + (size_t)n0 * DIN;
      const float* wp = wbase + (size_t)n0 * DIN * U_DIM;
      if (iter == 0) {
        // softmax of all-ones is exactly uniform: fold 1/N into the combine
        for (int nn = 0; nn < NPW; ++nn) {
          __builtin_prefetch(wp + 4 * DIN * U_DIM, 0, 1);
          v8f d = {0.f,0.f,0.f,0.f,0.f,0.f,0.f,0.f};
          d = uhatT_tile(xp, wp, d);
          #pragma unroll
          for (int j = 0; j < 8; ++j) sacc[j] += d[j];
          xp += DIN; wp += DIN * U_DIM;
        }
      } else {
        const float rsinv = rowsuminv[mcol];
        for (int nn = 0; nn < NPW; ++nn) {
          __builtin_prefetch(wp + 4 * DIN * U_DIM, 0, 1);
          v8f d = {0.f,0.f,0.f,0.f,0.f,0.f,0.f,0.f};
          d = uhatT_tile(xp, wp, d);
          float q = 0.0f;
          #pragma unroll
          for (int j = 0; j < 8; ++j) q = fmaf(d[j], vrow[j], q);
          q += __shfl_xor(q, 16);
          const float coeff = __expf(1.0f + q) * rsinv;
          #pragma unroll
          for (int j = 0; j < 8; ++j) sacc[j] = fmaf(coeff, d[j], sacc[j]);
          xp += DIN; wp += DIN * U_DIM;
        }
      }
    }
    const float scale0 = (iter == 0) ? (1.0f / (float)N_CAPS) : 1.0f;
    #pragma unroll
    for (int j = 0; j < 8; ++j)                       // store transposed [u][m]
      sred[(wv * BT + hi8 + j) * U_DIM + mcol] = sacc[j] * scale0;
    __syncthreads();

    // ---- combine across waves + squash ----
    {
      const int uu = tid >> 4, mm = tid & 15;         // one (u,m) per thread
      float s = 0.f;
      for (int w = 0; w < WAVES; ++w) s += sred[(w * BT + uu) * U_DIM + mm];
      svec[mm * U_DIM + uu] = s;                      // back to m-major
      __syncthreads();
      sred[mm * U_DIM + uu] = s * s;
      __syncthreads();
      if (tid < BT) {
        float sq = 0.f;
        for (int u2 = 0; u2 < U_DIM; ++u2) sq += sred[tid * U_DIM + u2];
        rowscale[tid] = sq / ((1.0f + sq) * sqrtf(sq + 1e-9f));
      }
      __syncthreads();
      const int mm2 = tid >> 4, uu2 = tid & 15;
      const float v = svec[mm2 * U_DIM + uu2] * rowscale[mm2];
      svec[mm2 * U_DIM + uu2] = v;
      vcum[mm2 * U_DIM + uu2] += v;
      __syncthreads();
    }
  }

  // ---- output: out[b, c, u] = v ----
  {
    const int mm = tid >> 4, uu = tid & 15;
    out[((size_t)(bt0 + mm) * C_CAPS + c) * U_DIM + uu] = svec[mm * U_DIM + uu];
  }
}

extern "C" void kernel_launch(void* const* d_in, const int* in_sizes, int n_in,
                              void* d_out, int out_size, void* d_ws, size_t ws_size,
                              hipStream_t stream) {
  (void)in_sizes; (void)n_in; (void)out_size; (void)d_ws; (void)ws_size;
  const float* X = (const float*)d_in[0];   // (256,1152,8) f32
  const float* W = (const float*)d_in[1];   // (10,1152,8,16) f32
  float* out = (float*)d_out;               // (256,10,16) f32
  dim3 grid(C_CAPS * (B_TOT / BT));         // 10 * 16 = 160 workgroups
  caps_route_kernel<<<grid, 256, 0, stream>>>(X, W, out);
}